// MultiresHashEncoding_84052509982984
// MI455X (gfx1250) — compile-verified
//
#include <hip/hip_runtime.h>

// ---- hyperparameters matching the reference ----
#define NLEVELS   16
#define LOG2_H    19
#define HASH_SIZE (1u << LOG2_H)
#define HASH_MASK (HASH_SIZE - 1u)
#define PRIME_Y   2654435761u
#define PRIME_Z   805459861u
#define FEATS_OUT (NLEVELS * 2)   // 32 floats per point

// resolutions = floor(16 * 2^(l/3)), l = 0..15 (matches np.floor(BASE*b**l))
__device__ __constant__ int kRes[NLEVELS] = {
    16, 20, 25, 32, 40, 50, 64, 80, 101, 128, 161, 203, 256, 322, 406, 512
};

// LDS tile: per wave, 32 points x 32 feats, stride 34 floats (conflict-free b64)
#define LDS_STRIDE 34

__launch_bounds__(256)
__global__ void MultiresHashEncoding_kernel(const float* __restrict__ xp,
                                            const float* __restrict__ tab,
                                            float* __restrict__ out,
                                            int npts)
{
    __shared__ float lds[8 * 32 * LDS_STRIDE];

    const int lane     = threadIdx.x & 31;
    const int wave     = threadIdx.x >> 5;
    const int waveBase = blockIdx.x * 256 + wave * 32;
    const int n        = waveBase + lane;

    float* wlds = &lds[wave * 32 * LDS_STRIDE];

    // ---- load coords (coalesced b96 pattern), prefetch ahead ----
    float fx = 0.f, fy = 0.f, fz = 0.f;
    if (n < npts) {
        fx = xp[3 * n + 0];
        fy = xp[3 * n + 1];
        fz = xp[3 * n + 2];
        int pn = n + 4096;
        if (pn >= npts) pn = npts - 1;
        __builtin_prefetch(xp + 3 * pn, 0, 1);   // -> global_prefetch_b8
    }

    // ---- 16 levels, fully unrolled so gathers pipeline under LOADcnt ----
#pragma unroll
    for (int l = 0; l < NLEVELS; ++l) {
        const float res = (float)kRes[l];

        const float sxf = fx * res, syf = fy * res, szf = fz * res;
        const float gx = floorf(sxf), gy = floorf(syf), gz = floorf(szf);
        const float tx = sxf - gx, ty = syf - gy, tz = szf - gz;

        const unsigned ix = (unsigned)gx, iy = (unsigned)gy, iz = (unsigned)gz;
        const unsigned hx0 = ix;            // prime for axis0 is 1
        const unsigned hx1 = ix + 1u;
        const unsigned hy0 = iy * PRIME_Y;
        const unsigned hy1 = hy0 + PRIME_Y;
        const unsigned hz0 = iz * PRIME_Z;
        const unsigned hz1 = hz0 + PRIME_Z;

        const unsigned e00 = hx0 ^ hy0, e10 = hx1 ^ hy0;
        const unsigned e01 = hx0 ^ hy1, e11 = hx1 ^ hy1;

        const unsigned h0 = (e00 ^ hz0) & HASH_MASK;  // (0,0,0)
        const unsigned h1 = (e10 ^ hz0) & HASH_MASK;  // (1,0,0)
        const unsigned h2 = (e01 ^ hz0) & HASH_MASK;  // (0,1,0)
        const unsigned h3 = (e11 ^ hz0) & HASH_MASK;  // (1,1,0)
        const unsigned h4 = (e00 ^ hz1) & HASH_MASK;  // (0,0,1)
        const unsigned h5 = (e10 ^ hz1) & HASH_MASK;  // (1,0,1)
        const unsigned h6 = (e01 ^ hz1) & HASH_MASK;  // (0,1,1)
        const unsigned h7 = (e11 ^ hz1) & HASH_MASK;  // (1,1,1)

        const float2* t = (const float2*)tab + (size_t)l * HASH_SIZE;

        // issue all 8 gathers (global_load_b64, random -> mostly L2 hits)
        const float2 c0 = t[h0];
        const float2 c1 = t[h1];
        const float2 c2 = t[h2];
        const float2 c3 = t[h3];
        const float2 c4 = t[h4];
        const float2 c5 = t[h5];
        const float2 c6 = t[h6];
        const float2 c7 = t[h7];

        const float sx = 1.f - tx, sy = 1.f - ty, sz = 1.f - tz;
        const float w00 = sx * sy, w10 = tx * sy;
        const float w01 = sx * ty, w11 = tx * ty;

        float a = 0.f, b = 0.f;
        float w;
        w = w00 * sz; a = fmaf(w, c0.x, a); b = fmaf(w, c0.y, b);
        w = w10 * sz; a = fmaf(w, c1.x, a); b = fmaf(w, c1.y, b);
        w = w01 * sz; a = fmaf(w, c2.x, a); b = fmaf(w, c2.y, b);
        w = w11 * sz; a = fmaf(w, c3.x, a); b = fmaf(w, c3.y, b);
        w = w00 * tz; a = fmaf(w, c4.x, a); b = fmaf(w, c4.y, b);
        w = w10 * tz; a = fmaf(w, c5.x, a); b = fmaf(w, c5.y, b);
        w = w01 * tz; a = fmaf(w, c6.x, a); b = fmaf(w, c6.y, b);
        w = w11 * tz; a = fmaf(w, c7.x, a); b = fmaf(w, c7.y, b);

        // stage into the wave-private LDS tile (ds_store_b64)
        float2 v; v.x = a; v.y = b;
        *(float2*)&wlds[lane * LDS_STRIDE + 2 * l] = v;
    }

    __syncthreads();

    // ---- coalesced writeback: wave covers its contiguous 4 KB output block ----
    // element e in [0,1024) of this wave's block: point = e/32, feat = e%32
#pragma unroll
    for (int r = 0; r < 16; ++r) {
        const int e     = r * 64 + lane * 2;
        const int point = e >> 5;
        const int feat  = e & 31;
        if (waveBase + point < npts) {
            float2 v = *(const float2*)&wlds[point * LDS_STRIDE + feat];
            *(float2*)&out[(size_t)waveBase * FEATS_OUT + e] = v;  // 256 B / wave-store
        }
    }
}

extern "C" void kernel_launch(void* const* d_in, const int* in_sizes, int n_in,
                              void* d_out, int out_size, void* d_ws, size_t ws_size,
                              hipStream_t stream) {
    const float* x      = (const float*)d_in[0];   // [N,3] float32
    const float* tables = (const float*)d_in[1];   // [16, 2^19, 2] float32
    float* out          = (float*)d_out;           // [N, 32] float32

    const int npts   = in_sizes[0] / 3;
    const int blocks = (npts + 255) / 256;
    MultiresHashEncoding_kernel<<<blocks, 256, 0, stream>>>(x, tables, out, npts);
}